// old_MambaBlock_59021440582436
// MI455X (gfx1250) — compile-verified
//
#include <hip/hip_runtime.h>
#include <hip/hip_bf16.h>
#include <math.h>

#define D_MODEL 768
#define D_STATE 16
#define D_CONV  4
#define D_INNER 1536
#define BATCH   2
#define SEQ     2048
#define ROWS    (BATCH * SEQ)          // 4096

// ---------------- WMMA fp32 GEMM:  C[M,N] = A[M,K] @ W[N,K]^T + bias ----------------
typedef float v2f __attribute__((ext_vector_type(2)));
typedef float v8f __attribute__((ext_vector_type(8)));
typedef unsigned int u32x4 __attribute__((ext_vector_type(4)));
typedef int i32x8 __attribute__((ext_vector_type(8)));
typedef int i32x4 __attribute__((ext_vector_type(4)));

#define BM 128
#define BN 128
#define BK 16
#define MT 2                 // 16-row m-tiles per wave (32 rows)
#define NT 4                 // 16-col n-tiles per wave (64 cols)
#define LDS_STRIDE 20        // BK + 4: 16B-aligned rows (TDM pad) + conflict-free frag reads

#if __has_builtin(__builtin_amdgcn_tensor_load_to_lds)
#define USE_TDM 1
#else
#define USE_TDM 0
#endif

#if USE_TDM
// Issue a TDM 2D tile load: tile = 128 rows x 16 f32, row stride K elements,
// LDS rows padded to 20 dwords via pad_interval=16dw / pad_amount=4dw.
__device__ __forceinline__ void tdm_load_tile(const float* gbase,
                                              unsigned lds_byte_addr,
                                              int total_rows, int K) {
    unsigned long long ga = (unsigned long long)(uintptr_t)gbase;
    u32x4 g0;
    g0[0] = 1u;                                        // count=1 (valid descriptor)
    g0[1] = lds_byte_addr;                             // lds_addr
    g0[2] = (unsigned)(ga & 0xFFFFFFFFull);            // global_addr[31:0]
    g0[3] = (unsigned)((ga >> 32) & 0x1FFFFFFull)      // global_addr[56:32]
          | (2u << 30);                                // type=2 (image)
    i32x8 g1;
    g1[0] = (2 << 16)        // data_size = 4 bytes
          | (1 << 20)        // pad_enable
          | (3 << 22)        // pad_interval: 16 dwords
          | (3 << 25);       // pad_amount: 4 dwords -> LDS row stride 20 dwords
    g1[1] = (int)(((unsigned)K & 0xFFFFu) << 16);                       // tensor_dim0[15:0]
    g1[2] = (int)(((unsigned)K >> 16) |
                  (((unsigned)total_rows & 0xFFFFu) << 16));            // dim0 hi | dim1 lo
    g1[3] = (int)((((unsigned)total_rows >> 16) & 0xFFFFu) |
                  (16u << 16));                                         // dim1 hi | tile_dim0=16
    g1[4] = (int)128u;                                                  // tile_dim1=128 rows
    g1[5] = (int)K;                                                     // dim0_stride[31:0]
    g1[6] = 0;                                                          // dim0_stride hi | dim1_stride lo
    g1[7] = 0;
    i32x4 z4 = {0, 0, 0, 0};
    i32x8 z8 = {0, 0, 0, 0, 0, 0, 0, 0};
    __builtin_amdgcn_tensor_load_to_lds(g0, g1, z4, z4, z8, 0);
}
#endif

__global__ __launch_bounds__(256)
void gemm_f32_wmma(const float* __restrict__ A,   // [M,K] row-major
                   const float* __restrict__ W,   // [N,K] row-major (we compute A @ W^T)
                   const float* __restrict__ bias,// [N]
                   float* __restrict__ C,         // [M,N]
                   int M, int N, int K)
{
    __shared__ float As[2][BM * LDS_STRIDE];
    __shared__ float Ws[2][BN * LDS_STRIDE];

    const int tid   = threadIdx.x;
    const int wave  = tid >> 5;          // 0..7
    const int lane  = tid & 31;
    const int l16   = lane & 15;
    const int khalf = (lane >> 4) * 2;   // K pair {0,1} or {2,3} per half-wave

    const int m_block = blockIdx.y * BM;
    const int n_block = blockIdx.x * BN;
    const int wave_m  = (wave & 3) * (MT * 16);   // 0,32,64,96
    const int wave_n  = (wave >> 2) * (NT * 16);  // 0,64

    v8f acc[MT * NT];
    #pragma unroll
    for (int i = 0; i < MT * NT; ++i)
        acc[i] = v8f{0.f, 0.f, 0.f, 0.f, 0.f, 0.f, 0.f, 0.f};

    const int nchunks = K / BK;

#if USE_TDM
    // ---- TDM double-buffered pipeline: one wave drives the DMA ----
    if (wave == 0) {
        tdm_load_tile(&A[(size_t)m_block * K], (unsigned)(uintptr_t)&As[0][0], M, K);
        tdm_load_tile(&W[(size_t)n_block * K], (unsigned)(uintptr_t)&Ws[0][0], N, K);
    }
    __builtin_amdgcn_s_wait_tensorcnt(0);
    __syncthreads();

    for (int c = 0; c < nchunks; ++c) {
        const int cur = c & 1;
        if (wave == 0 && (c + 1) < nchunks) {
            const int k1 = (c + 1) * BK;
            tdm_load_tile(&A[(size_t)m_block * K + k1],
                          (unsigned)(uintptr_t)&As[cur ^ 1][0], M, K);
            tdm_load_tile(&W[(size_t)n_block * K + k1],
                          (unsigned)(uintptr_t)&Ws[cur ^ 1][0], N, K);
        }
        const float* as = &As[cur][0];
        const float* ws = &Ws[cur][0];
        #pragma unroll
        for (int kk = 0; kk < BK; kk += 4) {
            v2f a[MT];
            #pragma unroll
            for (int mt = 0; mt < MT; ++mt) {
                const float* ap = &as[(wave_m + mt * 16 + l16) * LDS_STRIDE + kk + khalf];
                a[mt].x = ap[0]; a[mt].y = ap[1];
            }
            #pragma unroll
            for (int nt = 0; nt < NT; ++nt) {
                v2f b;
                const float* bp = &ws[(wave_n + nt * 16 + l16) * LDS_STRIDE + kk + khalf];
                b.x = bp[0]; b.y = bp[1];
                #pragma unroll
                for (int mt = 0; mt < MT; ++mt)
                    acc[mt * NT + nt] = __builtin_amdgcn_wmma_f32_16x16x4_f32(
                        false, a[mt], false, b, (short)0, acc[mt * NT + nt], false, false);
            }
        }
        __builtin_amdgcn_s_wait_tensorcnt(0);
        __syncthreads();
    }
#else
    // ---- fallback: synchronous VGPR staging, single buffer ----
    for (int c = 0; c < nchunks; ++c) {
        const int k0 = c * BK;
        #pragma unroll
        for (int i = 0; i < 2; ++i) {   // A tile: 128x16 = 512 float4
            int idx  = tid + i * 256;
            int row  = idx >> 2;
            int col4 = (idx & 3) * 4;
            const float4 v = *reinterpret_cast<const float4*>(
                &A[(size_t)(m_block + row) * K + k0 + col4]);
            float* dst = &As[0][row * LDS_STRIDE + col4];
            dst[0] = v.x; dst[1] = v.y; dst[2] = v.z; dst[3] = v.w;
        }
        #pragma unroll
        for (int i = 0; i < 2; ++i) {   // W tile: 128x16 = 512 float4
            int idx  = tid + i * 256;
            int row  = idx >> 2;
            int col4 = (idx & 3) * 4;
            const float4 v = *reinterpret_cast<const float4*>(
                &W[(size_t)(n_block + row) * K + k0 + col4]);
            float* dst = &Ws[0][row * LDS_STRIDE + col4];
            dst[0] = v.x; dst[1] = v.y; dst[2] = v.z; dst[3] = v.w;
        }
        __syncthreads();
        #pragma unroll
        for (int kk = 0; kk < BK; kk += 4) {
            v2f a[MT];
            #pragma unroll
            for (int mt = 0; mt < MT; ++mt) {
                const float* ap = &As[0][(wave_m + mt * 16 + l16) * LDS_STRIDE + kk + khalf];
                a[mt].x = ap[0]; a[mt].y = ap[1];
            }
            #pragma unroll
            for (int nt = 0; nt < NT; ++nt) {
                v2f b;
                const float* bp = &Ws[0][(wave_n + nt * 16 + l16) * LDS_STRIDE + kk + khalf];
                b.x = bp[0]; b.y = bp[1];
                #pragma unroll
                for (int mt = 0; mt < MT; ++mt)
                    acc[mt * NT + nt] = __builtin_amdgcn_wmma_f32_16x16x4_f32(
                        false, a[mt], false, b, (short)0, acc[mt * NT + nt], false, false);
            }
        }
        __syncthreads();
    }
#endif

    // epilogue: C/D layout: reg r -> row r (+8 for lanes 16-31), col = l16
    #pragma unroll
    for (int mt = 0; mt < MT; ++mt) {
        const int row_base = m_block + wave_m + mt * 16 + (lane >> 4) * 8;
        #pragma unroll
        for (int nt = 0; nt < NT; ++nt) {
            const int col = n_block + wave_n + nt * 16 + l16;
            const float bv = bias[col];
            #pragma unroll
            for (int r = 0; r < 8; ++r)
                C[(size_t)(row_base + r) * N + col] = acc[mt * NT + nt][r] + bv;
        }
    }
}

// ---------------- depthwise causal conv1d (k=4) + SiLU ----------------
__global__ __launch_bounds__(256)
void conv_silu_kernel(const float* __restrict__ xz,    // [ROWS, 2*D_INNER], x half at cols [0,D_INNER)
                      const float* __restrict__ convW, // [D_INNER,1,4]
                      const float* __restrict__ convb, // [D_INNER]
                      float* __restrict__ xc)          // [ROWS, D_INNER]
{
    int idx = blockIdx.x * 256 + threadIdx.x;
    if (idx >= ROWS * D_INNER) return;
    int d = idx % D_INNER;
    int t = (idx / D_INNER) % SEQ;
    int b = idx / (D_INNER * SEQ);
    float acc = convb[d];
    #pragma unroll
    for (int k = 0; k < D_CONV; ++k) {
        int ts = t - (D_CONV - 1) + k;
        if (ts >= 0) {
            float xv = xz[(size_t)(b * SEQ + ts) * (2 * D_INNER) + d];
            acc = fmaf(xv, convW[d * D_CONV + k], acc);
        }
    }
    xc[idx] = acc / (1.f + __expf(-acc));   // SiLU
}

// ---------------- x_proj: [ROWS,1536] @ xp_W[33,1536]^T -> B,C,delta_raw ----------------
__global__ __launch_bounds__(256)
void xproj_kernel(const float* __restrict__ xc,   // [ROWS, D_INNER]
                  const float* __restrict__ xpW,  // [33, D_INNER]
                  const float* __restrict__ xpb,  // [33]
                  float* __restrict__ Bm,         // [ROWS, 16]
                  float* __restrict__ Cm,         // [ROWS, 16]
                  float* __restrict__ draw)       // [ROWS]
{
    const int gwave = (blockIdx.x * blockDim.x + threadIdx.x) >> 5;
    const int lane  = threadIdx.x & 31;
    if (gwave >= ROWS) return;
    const float* row = xc + (size_t)gwave * D_INNER;

    float acc[2 * D_STATE + 1];
    #pragma unroll
    for (int n = 0; n < 2 * D_STATE + 1; ++n) acc[n] = 0.f;

    for (int k = lane; k < D_INNER; k += 32) {
        const float xv = row[k];
        #pragma unroll
        for (int n = 0; n < 2 * D_STATE + 1; ++n)
            acc[n] = fmaf(xv, xpW[n * D_INNER + k], acc[n]);
    }
    #pragma unroll
    for (int n = 0; n < 2 * D_STATE + 1; ++n) {
        float v = acc[n];
        #pragma unroll
        for (int off = 16; off >= 1; off >>= 1)
            v += __shfl_xor(v, off, 32);
        acc[n] = v;
    }
    if (lane == 0) {
        #pragma unroll
        for (int n = 0; n < D_STATE; ++n) {
            Bm[(size_t)gwave * D_STATE + n] = acc[n] + xpb[n];
            Cm[(size_t)gwave * D_STATE + n] = acc[D_STATE + n] + xpb[D_STATE + n];
        }
        draw[gwave] = acc[2 * D_STATE] + xpb[2 * D_STATE];
    }
}

// ---------------- delta = softplus(delta_raw * dp_W + dp_b) ----------------
__global__ __launch_bounds__(256)
void delta_kernel(const float* __restrict__ draw, // [ROWS]
                  const float* __restrict__ dpW,  // [D_INNER]
                  const float* __restrict__ dpb,  // [D_INNER]
                  float* __restrict__ delta)      // [ROWS, D_INNER]
{
    int idx = blockIdx.x * 256 + threadIdx.x;
    if (idx >= ROWS * D_INNER) return;
    int d   = idx % D_INNER;
    int row = idx / D_INNER;
    float x = fmaf(draw[row], dpW[d], dpb[d]);
    delta[idx] = (x > 20.f) ? x : log1pf(__expf(x));
}

// ---------------- selective scan: one lane per (b,d,n), half-wave reduction over n ----------------
__global__ __launch_bounds__(256)
void scan_kernel(const float* __restrict__ delta, // [ROWS, D_INNER]
                 const float* __restrict__ xc,    // [ROWS, D_INNER]
                 const float* __restrict__ Bm,    // [ROWS, 16]
                 const float* __restrict__ Cm,    // [ROWS, 16]
                 const float* __restrict__ A_log, // [D_INNER, 16]
                 float* __restrict__ y)           // [ROWS, D_INNER]
{
    const int tid = blockIdx.x * 256 + threadIdx.x;   // 0 .. BATCH*D_INNER*16-1
    const int n = tid & (D_STATE - 1);
    const int d = (tid >> 4) % D_INNER;
    const int b = tid / (D_INNER * D_STATE);

    const float Adn = -__expf(A_log[d * D_STATE + n]);
    float h = 0.f;

    const size_t row0 = (size_t)b * SEQ;
    for (int t = 0; t < SEQ; ++t) {
        const size_t r = row0 + t;
        const float dl = delta[r * D_INNER + d];   // broadcast within 16-lane group
        const float xv = xc[r * D_INNER + d];      // broadcast within 16-lane group
        const float Bv = Bm[r * D_STATE + n];
        const float Cv = Cm[r * D_STATE + n];
        const float Ab = __expf(dl * Adn);
        h = fmaf(Ab, h, dl * Bv * xv);
        float p = h * Cv;
        // reduce over the 16 states held by this half-wave
        p += __shfl_xor(p, 8, 16);
        p += __shfl_xor(p, 4, 16);
        p += __shfl_xor(p, 2, 16);
        p += __shfl_xor(p, 1, 16);
        if (n == 0) y[r * D_INNER + d] = p;
    }
}

// ---------------- gate: g = (y + xc*D) * silu(z) ----------------
__global__ __launch_bounds__(256)
void gate_kernel(const float* __restrict__ y,   // [ROWS, D_INNER]
                 const float* __restrict__ xc,  // [ROWS, D_INNER]
                 const float* __restrict__ xz,  // [ROWS, 2*D_INNER], z half at cols [D_INNER, 2*D_INNER)
                 const float* __restrict__ Dv,  // [D_INNER]
                 float* __restrict__ g)         // [ROWS, D_INNER]
{
    int idx = blockIdx.x * 256 + threadIdx.x;
    if (idx >= ROWS * D_INNER) return;
    int d   = idx % D_INNER;
    int row = idx / D_INNER;
    float z  = xz[(size_t)row * (2 * D_INNER) + D_INNER + d];
    float ys = fmaf(xc[idx], Dv[d], y[idx]);
    g[idx] = ys * (z / (1.f + __expf(-z)));
}

// ---------------- host launch ----------------
extern "C" void kernel_launch(void* const* d_in, const int* in_sizes, int n_in,
                              void* d_out, int out_size, void* d_ws, size_t ws_size,
                              hipStream_t stream) {
    const float* x      = (const float*)d_in[0];
    const float* in_W   = (const float*)d_in[1];
    const float* in_b   = (const float*)d_in[2];
    const float* conv_W = (const float*)d_in[3];
    const float* conv_b = (const float*)d_in[4];
    const float* xp_W   = (const float*)d_in[5];
    const float* xp_b   = (const float*)d_in[6];
    const float* dp_W   = (const float*)d_in[7];
    const float* dp_b   = (const float*)d_in[8];
    const float* A_log  = (const float*)d_in[9];
    const float* Dvec   = (const float*)d_in[10];
    const float* out_W  = (const float*)d_in[11];
    const float* out_b  = (const float*)d_in[12];
    float* out = (float*)d_out;

    // workspace carve-up (floats)
    float* ws    = (float*)d_ws;
    float* xz    = ws;                                 // ROWS * 3072
    float* xc    = xz   + (size_t)ROWS * 2 * D_INNER;  // ROWS * 1536
    float* Bm    = xc   + (size_t)ROWS * D_INNER;      // ROWS * 16
    float* Cm    = Bm   + (size_t)ROWS * D_STATE;      // ROWS * 16
    float* draw  = Cm   + (size_t)ROWS * D_STATE;      // ROWS
    float* delta = draw + (size_t)ROWS;                // ROWS * 1536
    float* yb    = delta + (size_t)ROWS * D_INNER;     // ROWS * 1536
    float* g     = delta;   // reuse: delta dead after scan, gate runs after scan

    // 1) in_proj: xz = x @ in_W^T + in_b      (M=4096, K=768, N=3072)
    {
        dim3 grid((2 * D_INNER) / BN, ROWS / BM);
        gemm_f32_wmma<<<grid, 256, 0, stream>>>(x, in_W, in_b, xz,
                                                ROWS, 2 * D_INNER, D_MODEL);
    }
    // 2) depthwise conv + SiLU
    {
        int n = ROWS * D_INNER;
        conv_silu_kernel<<<(n + 255) / 256, 256, 0, stream>>>(xz, conv_W, conv_b, xc);
    }
    // 3) x_proj -> B, C, delta_raw  (one wave per row)
    {
        int waves_per_block = 256 / 32;
        int blocks = (ROWS + waves_per_block - 1) / waves_per_block;
        xproj_kernel<<<blocks, 256, 0, stream>>>(xc, xp_W, xp_b, Bm, Cm, draw);
    }
    // 4) delta = softplus(...)
    {
        int n = ROWS * D_INNER;
        delta_kernel<<<(n + 255) / 256, 256, 0, stream>>>(draw, dp_W, dp_b, delta);
    }
    // 5) selective scan
    {
        int n = BATCH * D_INNER * D_STATE;   // 49152
        scan_kernel<<<n / 256, 256, 0, stream>>>(delta, xc, Bm, Cm, A_log, yb);
    }
    // 6) gate
    {
        int n = ROWS * D_INNER;
        gate_kernel<<<(n + 255) / 256, 256, 0, stream>>>(yb, xc, xz, Dvec, g);
    }
    // 7) out_proj: out = g @ out_W^T + out_b  (M=4096, K=1536, N=768)
    {
        dim3 grid(D_MODEL / BN, ROWS / BM);
        gemm_f32_wmma<<<grid, 256, 0, stream>>>(g, out_W, out_b, out,
                                                ROWS, D_MODEL, D_INNER);
    }
}